// GAT_55679956025853
// MI455X (gfx1250) — compile-verified
//
#include <hip/hip_runtime.h>

#define GAT_HEADS 8
#define NEG_SLOPE 0.2f
#define GAT_EPS 1e-16f

typedef float v2f __attribute__((ext_vector_type(2)));
typedef float v8f __attribute__((ext_vector_type(8)));

__device__ __forceinline__ float leaky(float v) { return v > 0.f ? v : NEG_SLOPE * v; }

// float atomic max via int punning; buffer must be initialized to -inf
__device__ __forceinline__ void atomicMaxF32(float* addr, float val) {
  if (val >= 0.f) atomicMax((int*)addr, __float_as_int(val));
  else            atomicMin((unsigned int*)addr, __float_as_uint(val));
}

__device__ __forceinline__ void edge_sd(const int* __restrict__ src,
                                        const int* __restrict__ dst,
                                        long e, long E, int& s, int& d) {
  if (e < E) { s = src[e]; d = dst[e]; }
  else       { s = d = (int)(e - E); }   // self loop
}

// ---------------------------------------------------------------------------
// C[M,Nc] = A[M,K] @ B[K,Nc], fp32 WMMA 16x16x4. One wave -> 16 rows x 64 cols.
// Requires K % 4 == 0, Nc % 64 == 0.
// ---------------------------------------------------------------------------
__global__ void gemm_wmma_f32(const float* __restrict__ A, const float* __restrict__ B,
                              float* __restrict__ C, int M, int K, int Nc) {
  const int lane   = threadIdx.x & 31;
  const int waveIn = threadIdx.x >> 5;
  const int wavesPerBlock = blockDim.x >> 5;

  const int colBlocks = Nc >> 6;            // 64-col blocks per row tile
  const int mTiles    = (M + 15) >> 4;
  const long totalWork = (long)mTiles * colBlocks;
  const long stride    = (long)gridDim.x * wavesPerBlock;

  const int mrow  = lane & 15;              // A row within tile, C col within 16-tile
  const int khalf = (lane >> 4) << 1;       // 0 (lanes 0-15) or 2 (lanes 16-31)

  for (long w = (long)blockIdx.x * wavesPerBlock + waveIn; w < totalWork; w += stride) {
    const int tm = (int)(w / colBlocks);
    const int cb = (int)(w % colBlocks);

    int row  = tm * 16 + mrow;
    int rowc = row < M ? row : (M - 1);     // clamp tail rows for loads
    const float* Arow = A + (size_t)rowc * K;

    v8f acc0 = {}, acc1 = {}, acc2 = {}, acc3 = {};

    for (int k = 0; k < K; k += 4) {
      // A fragment (16x4): lane (m, khalf), two consecutive K values
      v2f a;
      a.x = Arow[k + khalf];
      a.y = Arow[k + khalf + 1];

      const float* Bbase = B + (size_t)(k + khalf) * Nc + cb * 64 + mrow;
      // B fragments (4x16) for 4 column tiles
      v2f b0; b0.x = Bbase[0];  b0.y = Bbase[Nc];
      v2f b1; b1.x = Bbase[16]; b1.y = Bbase[Nc + 16];
      v2f b2; b2.x = Bbase[32]; b2.y = Bbase[Nc + 32];
      v2f b3; b3.x = Bbase[48]; b3.y = Bbase[Nc + 48];

      acc0 = __builtin_amdgcn_wmma_f32_16x16x4_f32(false, a, false, b0, (short)0, acc0, false, false);
      acc1 = __builtin_amdgcn_wmma_f32_16x16x4_f32(false, a, false, b1, (short)0, acc1, false, false);
      acc2 = __builtin_amdgcn_wmma_f32_16x16x4_f32(false, a, false, b2, (short)0, acc2, false, false);
      acc3 = __builtin_amdgcn_wmma_f32_16x16x4_f32(false, a, false, b3, (short)0, acc3, false, false);
    }

    // C/D layout: VGPR i, lanes 0-15 -> M=i; lanes 16-31 -> M=8+i; col = lane&15
    const int mbase = tm * 16 + ((lane >> 4) << 3);
    const int cbase = cb * 64 + mrow;
    v8f accs[4] = {acc0, acc1, acc2, acc3};
#pragma unroll
    for (int t = 0; t < 4; ++t) {
#pragma unroll
      for (int i = 0; i < 8; ++i) {
        int r = mbase + i;
        if (r < M) C[(size_t)r * Nc + cbase + t * 16] = accs[t][i];
      }
    }
  }
}

// ---------------------------------------------------------------------------
// alpha_s[n,h] = dot(h[n,h,:], a_src[h,:]) ; same for alpha_d. Thread per (n,h).
// ---------------------------------------------------------------------------
__global__ void alphas_kernel(const float* __restrict__ h, const float* __restrict__ a_src,
                              const float* __restrict__ a_dst, float* __restrict__ as_,
                              float* __restrict__ ad_, int n, int C) {
  int tid = blockIdx.x * blockDim.x + threadIdx.x;
  if (tid >= n * GAT_HEADS) return;
  int node = tid >> 3, hd = tid & 7;
  const float* hp = h + (size_t)node * GAT_HEADS * C + hd * C;
  float s = 0.f, d = 0.f;
  for (int c = 0; c < C; ++c) {
    float v = hp[c];
    s += v * a_src[hd * C + c];
    d += v * a_dst[hd * C + c];
  }
  as_[tid] = s;
  ad_[tid] = d;
}

__global__ void fill_kernel(float* __restrict__ p, float v, long n) {
  long i = (long)blockIdx.x * blockDim.x + threadIdx.x;
  if (i < n) p[i] = v;
}

__global__ void init_bias_kernel(float* __restrict__ out, const float* __restrict__ bias,
                                 long n, int F) {
  long i = (long)blockIdx.x * blockDim.x + threadIdx.x;
  if (i < n) out[i] = bias[i % F];
}

__global__ void relu_kernel(float* __restrict__ p, long n) {
  long i = (long)blockIdx.x * blockDim.x + threadIdx.x;
  if (i < n) { float v = p[i]; p[i] = v > 0.f ? v : 0.f; }
}

// thread per (edge, head): segment max of leaky(alpha_s[src]+alpha_d[dst])
__global__ void edge_max_kernel(const int* __restrict__ src, const int* __restrict__ dst,
                                const float* __restrict__ as_, const float* __restrict__ ad_,
                                float* __restrict__ maxb, long E, long Etot) {
  long tid = (long)blockIdx.x * blockDim.x + threadIdx.x;
  if (tid >= Etot * GAT_HEADS) return;
  long e = tid >> 3; int hd = (int)(tid & 7);
  int s, d; edge_sd(src, dst, e, E, s, d);
  float ev = leaky(as_[(size_t)s * GAT_HEADS + hd] + ad_[(size_t)d * GAT_HEADS + hd]);
  atomicMaxF32(&maxb[(size_t)d * GAT_HEADS + hd], ev);
}

// thread per (edge, head): segment sum of exp(e - max)
__global__ void edge_sum_kernel(const int* __restrict__ src, const int* __restrict__ dst,
                                const float* __restrict__ as_, const float* __restrict__ ad_,
                                const float* __restrict__ maxb, float* __restrict__ sumb,
                                long E, long Etot) {
  long tid = (long)blockIdx.x * blockDim.x + threadIdx.x;
  if (tid >= Etot * GAT_HEADS) return;
  long e = tid >> 3; int hd = (int)(tid & 7);
  int s, d; edge_sd(src, dst, e, E, s, d);
  size_t di = (size_t)d * GAT_HEADS + hd;
  float ev = leaky(as_[(size_t)s * GAT_HEADS + hd] + ad_[di]);
  atomicAdd(&sumb[di], __expf(ev - maxb[di]));
}

// one wave per edge: lane owns F/32 contiguous features (one head slice)
template <int F, int C>
__global__ void edge_scatter_kernel(const int* __restrict__ src, const int* __restrict__ dst,
                                    const float* __restrict__ as_, const float* __restrict__ ad_,
                                    const float* __restrict__ maxb, const float* __restrict__ sumb,
                                    const float* __restrict__ h, float* __restrict__ agg,
                                    long E, long Etot) {
  const int lane = threadIdx.x & 31;
  long wave = ((long)blockIdx.x * blockDim.x + threadIdx.x) >> 5;
  if (wave >= Etot) return;
  int s, d; edge_sd(src, dst, wave, E, s, d);

  constexpr int FPL = F / 32;          // features per lane (4 or 2)
  const int f0 = lane * FPL;
  const int hd = f0 / C;

  size_t di = (size_t)d * GAT_HEADS + hd;
  float ev    = leaky(as_[(size_t)s * GAT_HEADS + hd] + ad_[di]);
  float p     = __expf(ev - maxb[di]);
  float alpha = p / (sumb[di] + GAT_EPS);

  const float* hs = h + (size_t)s * F + f0;
  float* ao = agg + (size_t)d * F + f0;
#pragma unroll
  for (int i = 0; i < FPL; ++i) atomicAdd(&ao[i], hs[i] * alpha);
}

// ---------------------------------------------------------------------------
extern "C" void kernel_launch(void* const* d_in, const int* in_sizes, int n_in,
                              void* d_out, int out_size, void* d_ws, size_t ws_size,
                              hipStream_t stream) {
  const float* x    = (const float*)d_in[0];
  const int*   ei   = (const int*)d_in[1];
  // d_in[2] edge_weights: unused by reference
  const float* W1   = (const float*)d_in[3];
  const float* a_s1 = (const float*)d_in[4];
  const float* a_d1 = (const float*)d_in[5];
  const float* b1   = (const float*)d_in[6];
  const float* W2   = (const float*)d_in[7];
  const float* a_s2 = (const float*)d_in[8];
  const float* a_d2 = (const float*)d_in[9];
  const float* b2   = (const float*)d_in[10];
  float* out = (float*)d_out;

  const long N = in_sizes[0] / 128;
  const long E = in_sizes[1] / 2;
  const long Etot = E + N;
  const int* src = ei;
  const int* dst = ei + E;

  // workspace layout (floats)
  float* ws   = (float*)d_ws;
  float* h1   = ws;                 // N*128
  float* agg1 = h1 + N * 128;       // N*128 (relu'd in place -> layer-2 input)
  float* h2   = agg1 + N * 128;     // N*64
  float* as1  = h2 + N * 64;        // N*8
  float* ad1  = as1 + N * 8;
  float* as2  = ad1 + N * 8;
  float* ad2  = as2 + N * 8;
  float* maxb = ad2 + N * 8;        // N*8, reused across layers
  float* sumb = maxb + N * 8;       // N*8

  const int BLK = 256;
  const int wavesPerBlock = BLK / 32;
  const long nh   = N * GAT_HEADS;
  const long ehTot = Etot * GAT_HEADS;
  const long scatterThreads = Etot * 32;
  const float NEG_INF = -__builtin_inff();

  auto blocks = [](long work, int blk) { return (unsigned)((work + blk - 1) / blk); };

  // ---------------- layer 1 ----------------
  {
    long work = ((N + 15) / 16) * (128 / 64);
    gemm_wmma_f32<<<blocks(work, wavesPerBlock), BLK, 0, stream>>>(x, W1, h1, (int)N, 128, 128);
  }
  alphas_kernel<<<blocks(nh, BLK), BLK, 0, stream>>>(h1, a_s1, a_d1, as1, ad1, (int)N, 16);
  fill_kernel<<<blocks(nh, BLK), BLK, 0, stream>>>(maxb, NEG_INF, nh);
  fill_kernel<<<blocks(nh, BLK), BLK, 0, stream>>>(sumb, 0.f, nh);
  init_bias_kernel<<<blocks(N * 128, BLK), BLK, 0, stream>>>(agg1, b1, N * 128, 128);
  edge_max_kernel<<<blocks(ehTot, BLK), BLK, 0, stream>>>(src, dst, as1, ad1, maxb, E, Etot);
  edge_sum_kernel<<<blocks(ehTot, BLK), BLK, 0, stream>>>(src, dst, as1, ad1, maxb, sumb, E, Etot);
  edge_scatter_kernel<128, 16><<<blocks(scatterThreads, BLK), BLK, 0, stream>>>(
      src, dst, as1, ad1, maxb, sumb, h1, agg1, E, Etot);
  relu_kernel<<<blocks(N * 128, BLK), BLK, 0, stream>>>(agg1, N * 128);

  // ---------------- layer 2 ----------------
  {
    long work = ((N + 15) / 16) * (64 / 64);
    gemm_wmma_f32<<<blocks(work, wavesPerBlock), BLK, 0, stream>>>(agg1, W2, h2, (int)N, 128, 64);
  }
  alphas_kernel<<<blocks(nh, BLK), BLK, 0, stream>>>(h2, a_s2, a_d2, as2, ad2, (int)N, 8);
  fill_kernel<<<blocks(nh, BLK), BLK, 0, stream>>>(maxb, NEG_INF, nh);
  fill_kernel<<<blocks(nh, BLK), BLK, 0, stream>>>(sumb, 0.f, nh);
  init_bias_kernel<<<blocks(N * 64, BLK), BLK, 0, stream>>>(out, b2, N * 64, 64);
  edge_max_kernel<<<blocks(ehTot, BLK), BLK, 0, stream>>>(src, dst, as2, ad2, maxb, E, Etot);
  edge_sum_kernel<<<blocks(ehTot, BLK), BLK, 0, stream>>>(src, dst, as2, ad2, maxb, sumb, E, Etot);
  edge_scatter_kernel<64, 8><<<blocks(scatterThreads, BLK), BLK, 0, stream>>>(
      src, dst, as2, ad2, maxb, sumb, h2, out, E, Etot);
}